// DetectionLoss_13426067768049
// MI455X (gfx1250) — compile-verified
//
#include <hip/hip_runtime.h>
#include <hip/hip_bf16.h>

// ---------------- problem constants (match reference shapes) ----------------
#define B_IMG   16
#define M_SLOTS 8
#define H_DIM   112
#define W_DIM   112
#define NUM_CLS 64
#define N_GT    20
#define POS_W   10.0f
#define OBJ_ELEMS (B_IMG * M_SLOTS * H_DIM * W_DIM)   // 1,605,632
#define OBJ_CHUNKS (OBJ_ELEMS / 64)                   // 25,088 chunks of 64 f32

// reduction geometry: 112 blocks x 256 threads = 896 waves, 28 chunks/wave
#define RED_BLOCKS  112
#define RED_THREADS 256
#define RED_WAVES   (RED_BLOCKS * RED_THREADS / 32)   // 896
#define CHUNKS_PER_WAVE (OBJ_CHUNKS / RED_WAVES)      // 28
static_assert(OBJ_CHUNKS % RED_WAVES == 0, "exact tiling required (EXEC must stay all-ones for WMMA)");

typedef __attribute__((ext_vector_type(2))) float v2f;
typedef __attribute__((ext_vector_type(8))) float v8f;

__device__ __forceinline__ float safelog(float x) { return fmaxf(logf(x), -100.0f); }

// ws accumulators: acc[0]=obj_sum, acc[1]=box_sum, acc[2]=cls_sum
__global__ void dl_init_kernel(float* acc) {
    if (threadIdx.x < 4) acc[threadIdx.x] = 0.0f;
}

// ---------------------------------------------------------------------------
// Per-image sparse work: assignment, obj BCE corrections, GIoU, focal cls.
// 16 blocks (one per image), 32 threads (threads 0..19 own one GT each).
// ---------------------------------------------------------------------------
__global__ void dl_per_image_kernel(const float* __restrict__ obj,
                                    const float* __restrict__ boxes,
                                    const float* __restrict__ cls,
                                    const float* __restrict__ gtb,
                                    const int*   __restrict__ gtl,
                                    float* __restrict__ acc) {
    const int b = blockIdx.x;
    const int t = threadIdx.x;
    __shared__ int s_cx[N_GT], s_cy[N_GT], s_cell[N_GT];

    if (t < N_GT) {
        const float gx = gtb[(b * N_GT + t) * 4 + 0];
        const float gy = gtb[(b * N_GT + t) * 4 + 1];
        int cx = (int)(gx * (float)W_DIM); cx = min(max(cx, 0), W_DIM - 1);
        int cy = (int)(gy * (float)H_DIM); cy = min(max(cy, 0), H_DIM - 1);
        s_cx[t] = cx; s_cy[t] = cy; s_cell[t] = cy * W_DIM + cx;
    }
    __syncthreads();

    if (t < N_GT) {
        const int cx = s_cx[t], cy = s_cy[t], cell = s_cell[t];

        // rank = #earlier GTs in same cell (tril(eq).sum)
        int rank = 0;
        for (int j = 0; j < t; ++j) rank += (s_cell[j] == cell) ? 1 : 0;
        const bool valid  = rank < M_SLOTS;
        const int  slot_t = min(rank, M_SLOTS - 1);

        // objectness BCE correction at the positive cell (unique per image)
        if (valid) {
            const float p = obj[((b * M_SLOTS + slot_t) * H_DIM + cy) * W_DIM + cx];
            atomicAdd(&acc[0], -POS_W * safelog(p) + safelog(1.0f - p));
        }

        // slot = first m with obj score > 0.5, else 0 (argmax of bool)
        int slot = 0;
        for (int m = M_SLOTS - 1; m >= 0; --m) {
            const float p = obj[((b * M_SLOTS + m) * H_DIM + cy) * W_DIM + cx];
            if (p > 0.5f) slot = m;
        }

        // -------- GIoU box loss --------
        float bx[4], g4[4];
        for (int k = 0; k < 4; ++k) {
            bx[k] = boxes[(((b * M_SLOTS + slot) * 4 + k) * H_DIM + cy) * W_DIM + cx];
            g4[k] = gtb[(b * N_GT + t) * 4 + k];
        }
        const float p0 = bx[0] - 0.5f * bx[2], p1 = bx[1] - 0.5f * bx[3];
        const float p2 = bx[0] + 0.5f * bx[2], p3 = bx[1] + 0.5f * bx[3];
        const float t0 = g4[0] - 0.5f * g4[2], t1 = g4[1] - 0.5f * g4[3];
        const float t2 = g4[0] + 0.5f * g4[2], t3 = g4[1] + 0.5f * g4[3];
        const float px1 = fminf(p0, p2), py1 = fminf(p1, p3);
        const float px2 = fmaxf(p0, p2), py2 = fmaxf(p1, p3);
        const float ix1 = fmaxf(px1, t0), iy1 = fmaxf(py1, t1);
        const float ix2 = fminf(px2, t2), iy2 = fminf(py2, t3);
        const float inter = fmaxf(ix2 - ix1, 0.0f) * fmaxf(iy2 - iy1, 0.0f);
        const float pa  = (px2 - px1) * (py2 - py1);
        const float ta  = (t2 - t0) * (t3 - t1);
        const float uni = pa + ta - inter;
        const float iou = inter / (uni + 1e-6f);
        const float ex1 = fminf(px1, t0), ey1 = fminf(py1, t1);
        const float ex2 = fmaxf(px2, t2), ey2 = fmaxf(py2, t3);
        const float enc = fmaxf(ex2 - ex1, 0.0f) * fmaxf(ey2 - ey1, 0.0f);
        float giou = iou - (enc - uni) / (enc + 1e-6f);
        giou = fminf(fmaxf(giou, -1.0f), 1.0f);
        atomicAdd(&acc[1], fmaxf(1.0f - giou, 0.0f));

        // -------- focal classification loss --------
        const int lbl = gtl[b * N_GT + t];
        const size_t hw = (size_t)H_DIM * W_DIM;
        const float* lg = cls + ((size_t)(b * M_SLOTS + slot) * NUM_CLS * H_DIM + cy) * W_DIM + cx;
        float mx = -1e30f;
        for (int c = 0; c < NUM_CLS; ++c) mx = fmaxf(mx, lg[(size_t)c * hw]);
        float se = 0.0f;
        for (int c = 0; c < NUM_CLS; ++c) se += expf(lg[(size_t)c * hw] - mx);
        const float logp = lg[(size_t)lbl * hw] - mx - logf(se);
        const float ce = -logp;
        const float pt = expf(logp) + 1e-7f;
        const float alpha = (lbl == 0) ? 0.25f : 0.75f;
        const float om = 1.0f - pt;
        atomicAdd(&acc[2], alpha * om * om * ce);
    }
}

// ---------------------------------------------------------------------------
// Dense objectness reduction: sum of -safelog(1-p) over all elements, using
// V_WMMA_F32_16X16X4_F32 as an f32 reduction engine.
//   A (16x4, 2 VGPRs) <- 64 transformed values per wave-step
//   B (4x16)          <- all ones  =>  D[m][n] = row-sum of A (chained into C)
// C layout: lane L(0..15) col N=L, VGPR v = row M=v (lanes 16..31: M=v+8).
// Sum of a lane's 8 C regs => half-wave total in lanes 0 and 16; compiler
// folds the two-lane finish into a readlane sum + one global_atomic_add_f32.
// ---------------------------------------------------------------------------
__global__ void dl_obj_reduce_kernel(const float* __restrict__ obj,
                                     float* __restrict__ acc) {
    const int wave = (blockIdx.x * RED_THREADS + threadIdx.x) >> 5;
    const int lane = threadIdx.x & 31;
    const float2* p2 = (const float2*)(obj + (size_t)wave * CHUNKS_PER_WAVE * 64);

    v8f c = {0.f, 0.f, 0.f, 0.f, 0.f, 0.f, 0.f, 0.f};
    v2f ones; ones.x = 1.0f; ones.y = 1.0f;

    for (int i = 0; i < CHUNKS_PER_WAVE; ++i) {
        // prefetch ~8 chunks ahead, clamped to this wave's own range so we
        // never touch pages past the end of the allocation (branchless, so
        // EXEC stays all-ones for the WMMA below)
        const int pf = min(i + 8, CHUNKS_PER_WAVE - 1);
        __builtin_prefetch(p2 + pf * 32 + lane, 0, 3);  // global_prefetch_b8
        const float2 v = p2[i * 32 + lane];
        v2f a;
        a.x = -fmaxf(logf(1.0f - v.x), -100.0f);   // -safelog(1-p), target=0 BCE
        a.y = -fmaxf(logf(1.0f - v.y), -100.0f);
        c = __builtin_amdgcn_wmma_f32_16x16x4_f32(
                /*neg_a=*/false, a, /*neg_b=*/false, ones,
                /*c_mod=*/(short)0, c, /*reuse_a=*/false, /*reuse_b=*/false);
    }
    const float s = c[0] + c[1] + c[2] + c[3] + c[4] + c[5] + c[6] + c[7];
    if (lane == 0 || lane == 16) atomicAdd(&acc[0], s);
}

__global__ void dl_finalize_kernel(const float* __restrict__ acc,
                                   float* __restrict__ out) {
    if (threadIdx.x == 0) {
        out[0] = acc[1] * (1.0f / (float)(B_IMG * N_GT));          // box_loss * 1.0
        out[1] = acc[2] * (1.0f / (float)(B_IMG * N_GT));          // cls_loss * 1.0
        out[2] = acc[0] * (0.1f / (float)OBJ_ELEMS);               // obj_loss * 0.1
    }
}

extern "C" void kernel_launch(void* const* d_in, const int* in_sizes, int n_in,
                              void* d_out, int out_size, void* d_ws, size_t ws_size,
                              hipStream_t stream) {
    const float* obj   = (const float*)d_in[0];   // (B,M,H,W) f32
    const float* boxes = (const float*)d_in[1];   // (B,M,4,H,W) f32
    const float* cls   = (const float*)d_in[2];   // (B,M,C,H,W) f32
    const float* gtb   = (const float*)d_in[3];   // (B,N,4) f32
    const int*   gtl   = (const int*)d_in[4];     // (B,N) i32
    float* acc = (float*)d_ws;
    float* out = (float*)d_out;

    dl_init_kernel<<<1, 32, 0, stream>>>(acc);
    dl_per_image_kernel<<<B_IMG, 32, 0, stream>>>(obj, boxes, cls, gtb, gtl, acc);
    dl_obj_reduce_kernel<<<RED_BLOCKS, RED_THREADS, 0, stream>>>(obj, acc);
    dl_finalize_kernel<<<1, 1, 0, stream>>>(acc, out);
}